// ShapeSampler_26560077758934
// MI455X (gfx1250) — compile-verified
//
#include <hip/hip_runtime.h>

typedef float v2f __attribute__((ext_vector_type(2)));
typedef float v8f __attribute__((ext_vector_type(8)));

static constexpr int NF   = 16384;   // faces
static constexpr int NP   = 16384;   // points
// workspace layout (floats):
//   [0,3F)   ab xyz (SoA)     [3F,6F)  ac xyz     [6F,9F)  a xyz    [9F,12F) nrm xyz
//   [12F,23F) 11 per-face constants: ab.a ab.b ab.c ac.a ac.b ac.c |ab|2 |ac|2 ab.ac |a|2 a.nrm
//   [23F, 23F+NP) pp2 = |p|^2

__device__ __forceinline__ float sdiv_fast(float n, float d) {
    d = (d == 0.0f) ? 1.0f : d;
    return n * __builtin_amdgcn_rcpf(d);
}

__device__ __forceinline__ v8f wmma4(v2f a, v2f b) {
    v8f c = {};
    return __builtin_amdgcn_wmma_f32_16x16x4_f32(
        /*neg_a=*/false, a, /*neg_b=*/false, b,
        /*c_mod=*/(short)0, c, /*reuse_a=*/false, /*reuse_b=*/false);
}

__global__ __launch_bounds__(256) void sdf_pre(
    const float* __restrict__ verts, const int* __restrict__ faces,
    const float* __restrict__ pts, float* __restrict__ ws) {
    int t = blockIdx.x * blockDim.x + threadIdx.x;
    if (t < NF) {
        int i0 = faces[3 * t + 0], i1 = faces[3 * t + 1], i2 = faces[3 * t + 2];
        float ax = verts[3 * i0 + 0], ay = verts[3 * i0 + 1], az = verts[3 * i0 + 2];
        float bx = verts[3 * i1 + 0], by = verts[3 * i1 + 1], bz = verts[3 * i1 + 2];
        float cx = verts[3 * i2 + 0], cy = verts[3 * i2 + 1], cz = verts[3 * i2 + 2];
        float abx = bx - ax, aby = by - ay, abz = bz - az;
        float acx = cx - ax, acy = cy - ay, acz = cz - az;
        float nx = aby * acz - abz * acy;
        float ny = abz * acx - abx * acz;
        float nz = abx * acy - aby * acx;
        ws[0 * NF + t] = abx; ws[1 * NF + t] = aby; ws[2 * NF + t] = abz;
        ws[3 * NF + t] = acx; ws[4 * NF + t] = acy; ws[5 * NF + t] = acz;
        ws[6 * NF + t] = ax;  ws[7 * NF + t] = ay;  ws[8 * NF + t] = az;
        ws[9 * NF + t] = nx;  ws[10 * NF + t] = ny; ws[11 * NF + t] = nz;
        float* C = ws + 12 * NF;
        C[0 * NF + t]  = abx * ax + aby * ay + abz * az;   // ab.a
        C[1 * NF + t]  = abx * bx + aby * by + abz * bz;   // ab.b
        C[2 * NF + t]  = abx * cx + aby * cy + abz * cz;   // ab.c
        C[3 * NF + t]  = acx * ax + acy * ay + acz * az;   // ac.a
        C[4 * NF + t]  = acx * bx + acy * by + acz * bz;   // ac.b
        C[5 * NF + t]  = acx * cx + acy * cy + acz * cz;   // ac.c
        C[6 * NF + t]  = abx * abx + aby * aby + abz * abz; // |ab|^2
        C[7 * NF + t]  = acx * acx + acy * acy + acz * acz; // |ac|^2
        C[8 * NF + t]  = abx * acx + aby * acy + abz * acz; // ab.ac
        C[9 * NF + t]  = ax * ax + ay * ay + az * az;       // |a|^2
        C[10 * NF + t] = ax * nx + ay * ny + az * nz;       // a.nrm
    } else {
        int i = t - NF;
        if (i < NP) {
            float x = pts[3 * i + 0], y = pts[3 * i + 1], z = pts[3 * i + 2];
            ws[23 * NF + i] = x * x + y * y + z * z;
        }
    }
}

__global__ __launch_bounds__(256) void sdf_main(
    const float* __restrict__ pts, const float* __restrict__ ws,
    float* __restrict__ out) {
    __shared__ unsigned lds[8 * 16];

    const int tid  = threadIdx.x;
    const int lane = tid & 31;
    const int wv   = tid >> 5;        // wave id 0..7
    const int col  = lane & 15;       // B column / point row index
    const int hi   = lane >> 4;       // half-wave selector (K 0,1 vs 2,3)
    const int ptBase = blockIdx.x * 16;

    // --- A matrix: 16x4 f32, row m = point (ptBase+col), K = {x,y,z,0} ---
    const float* p = pts + 3 * (ptBase + col);
    float px = p[0], py = p[1], pz = p[2];
    v2f A;
    A[0] = hi ? pz : px;              // K=0 (lo half) / K=2 (hi half)
    A[1] = hi ? 0.0f : py;            // K=1 (lo half) / K=3 (hi half)

    // pp2 for the 8 C-matrix rows this lane owns: m = 8*hi + r
    const float* pp2p = ws + 23 * NF + ptBase + 8 * hi;
    float pp2v[8];
#pragma unroll
    for (int r = 0; r < 8; ++r) pp2v[r] = pp2p[r];

    const float* AB = ws + 0 * NF;
    const float* AC = ws + 3 * NF;
    const float* AV = ws + 6 * NF;
    const float* NM = ws + 9 * NF;
    const float* CST = ws + 12 * NF;
    const int xo = hi ? 2 * NF : 0;   // select x-plane (lo) or z-plane (hi)

    unsigned best[8];
#pragma unroll
    for (int r = 0; r < 8; ++r) best[r] = 0xFFFFFFFFu;

    for (int ch = wv; ch < NF / 16; ch += 8) {
        const int f = ch * 16 + col;

        v2f Bab, Bac, Bav, Bnm;
        Bab[0] = AB[xo + f]; Bab[1] = hi ? 0.0f : AB[NF + f];
        Bac[0] = AC[xo + f]; Bac[1] = hi ? 0.0f : AC[NF + f];
        Bav[0] = AV[xo + f]; Bav[1] = hi ? 0.0f : AV[NF + f];
        Bnm[0] = NM[xo + f]; Bnm[1] = hi ? 0.0f : NM[NF + f];

        float cab_a = CST[0 * NF + f], cab_b = CST[1 * NF + f], cab_c = CST[2 * NF + f];
        float cac_a = CST[3 * NF + f], cac_b = CST[4 * NF + f], cac_c = CST[5 * NF + f];
        float L1 = CST[6 * NF + f], L2 = CST[7 * NF + f], Mm = CST[8 * NF + f];
        float aa2 = CST[9 * NF + f], anc = CST[10 * NF + f];

        // 16x16 tiles of: ab.p, ac.p, a.p, nrm.p
        v8f S1 = wmma4(A, Bab);
        v8f S2 = wmma4(A, Bac);
        v8f S3 = wmma4(A, Bav);
        v8f S4 = wmma4(A, Bnm);

#pragma unroll
        for (int r = 0; r < 8; ++r) {
            float s1 = S1[r], s2 = S2[r], s3 = S3[r], s4 = S4[r];
            float d1 = s1 - cab_a, d3 = s1 - cab_b, d5 = s1 - cab_c;
            float d2 = s2 - cac_a, d4 = s2 - cac_b, d6 = s2 - cac_c;

            float va = d3 * d6 - d5 * d4;
            float vb = d5 * d2 - d1 * d6;
            float vc = d1 * d4 - d3 * d2;
            float denom = va + vb + vc;
            float v = sdiv_fast(vb, denom);
            float w = sdiv_fast(vc, denom);

            float e43 = d4 - d3, e56 = d5 - d6;
            float t_bc = sdiv_fast(e43, e43 + e56);
            bool c_bc = (va <= 0.0f) && (e43 >= 0.0f) && (e56 >= 0.0f);
            v = c_bc ? (1.0f - t_bc) : v;  w = c_bc ? t_bc : w;

            float t_ac = sdiv_fast(d2, d2 - d6);
            bool c_ac = (vb <= 0.0f) && (d2 >= 0.0f) && (d6 <= 0.0f);
            v = c_ac ? 0.0f : v;  w = c_ac ? t_ac : w;

            float t_ab = sdiv_fast(d1, d1 - d3);
            bool c_ab = (vc <= 0.0f) && (d1 >= 0.0f) && (d3 <= 0.0f);
            v = c_ab ? t_ab : v;  w = c_ab ? 0.0f : w;

            bool c_c = (d6 >= 0.0f) && (d5 <= d6);
            v = c_c ? 0.0f : v;  w = c_c ? 1.0f : w;
            bool c_b = (d3 >= 0.0f) && (d4 <= d3);
            v = c_b ? 1.0f : v;  w = c_b ? 0.0f : w;
            bool c_a = (d1 <= 0.0f) && (d2 <= 0.0f);
            v = c_a ? 0.0f : v;  w = c_a ? 0.0f : w;

            // dist2 = |p-a|^2 - 2(v d1 + w d2) + v^2 L1 + 2vw M + w^2 L2
            float ap2 = pp2v[r] - 2.0f * s3 + aa2;
            float dq = ap2 - 2.0f * (v * d1 + w * d2)
                     + v * (v * L1 + 2.0f * w * Mm) + w * w * L2;
            dq = fmaxf(dq, 0.0f);

            bool outward = (s4 - anc) > 0.0f;
            unsigned key = (__float_as_uint(dq) << 1) | (outward ? 1u : 0u);
            best[r] = best[r] < key ? best[r] : key;
        }
    }

    // reduce across the 16 columns (stays inside each 16-lane half)
#pragma unroll
    for (int off = 1; off < 16; off <<= 1) {
#pragma unroll
        for (int r = 0; r < 8; ++r) {
            unsigned o = (unsigned)__shfl_xor((int)best[r], off, 32);
            best[r] = best[r] < o ? best[r] : o;
        }
    }
    if (col == 0) {  // lanes 0 and 16
#pragma unroll
        for (int r = 0; r < 8; ++r) lds[wv * 16 + 8 * hi + r] = best[r];
    }
    __syncthreads();

    if (tid < 16) {
        unsigned k = 0xFFFFFFFFu;
#pragma unroll
        for (int w = 0; w < 8; ++w) {
            unsigned o = lds[w * 16 + tid];
            k = k < o ? k : o;
        }
        float d2min = __uint_as_float(k >> 1);
        float dist = sqrtf(fmaxf(d2min, 1e-12f));
        out[ptBase + tid] = (k & 1u) ? -dist : dist;
    }
}

extern "C" void kernel_launch(void* const* d_in, const int* in_sizes, int n_in,
                              void* d_out, int out_size, void* d_ws, size_t ws_size,
                              hipStream_t stream) {
    const float* pts   = (const float*)d_in[0];
    const float* verts = (const float*)d_in[1];
    const int*   faces = (const int*)d_in[2];
    float* out = (float*)d_out;
    float* ws  = (float*)d_ws;

    int preThreads = NF + NP;
    sdf_pre<<<(preThreads + 255) / 256, 256, 0, stream>>>(verts, faces, pts, ws);
    sdf_main<<<NP / 16, 256, 0, stream>>>(pts, ws, out);
}